// adpt_operation_88029649699404
// MI455X (gfx1250) — compile-verified
//
#include <hip/hip_runtime.h>
#include <hip/hip_bf16.h>

// ---------------------------------------------------------------------------
// Problem constants (from the reference)
// ---------------------------------------------------------------------------
#define BB   16
#define NN   4096
#define SS   512
#define KK   32
#define CINC 64
#define EE   64
#define COUTC 128

typedef float v2f __attribute__((ext_vector_type(2)));
typedef float v8f __attribute__((ext_vector_type(8)));

// ---------------------------------------------------------------------------
// Generic strided-batched WMMA GEMM:  C[b] = alpha * A[b] x B[b] (+ bias)
// One wave32 per 16x64 output strip: 4 accumulators share one A fragment per
// k-step (4x fewer A fetches and 4x less address math per WMMA).
// A indexed A[m*Asm + k*Ask], B indexed B[k*Bsk + n*Bsn], C[m*Csm + n*Csn].
// biasMode: 0 = none, 1 = bias[m], 2 = bias[n].
// Requires M mult of 16, N mult of 64, K mult of 4 (true for every call here).
// ---------------------------------------------------------------------------
__global__ void k_gemm(const float* __restrict__ A, const float* __restrict__ Bm,
                       const float* __restrict__ bias, float* __restrict__ C,
                       int M, int N, int K,
                       long long Abat, long long Asm, long long Ask,
                       long long Bbat, long long Bsk, long long Bsn,
                       long long Cbat, long long Csm, long long Csn,
                       int biasMode, float alpha)
{
    const float* Ab = A  + (long long)blockIdx.z * Abat;
    const float* Bb = Bm + (long long)blockIdx.z * Bbat;
    float*       Cb = C  + (long long)blockIdx.z * Cbat;

    const int n0   = blockIdx.x * 64;
    const int m0   = blockIdx.y * 16;
    const int lane = threadIdx.x & 31;
    const int half = lane >> 4;     // 0: K pair {0,1}, 1: K pair {2,3}
    const int l16  = lane & 15;

    v8f acc0 = {0.f,0.f,0.f,0.f,0.f,0.f,0.f,0.f};
    v8f acc1 = acc0, acc2 = acc0, acc3 = acc0;

    // per-lane base pointers at k = 2*half
    const long long k0off = (long long)(2 * half) * Ask;
    const float* ap  = Ab + (long long)(m0 + l16) * Asm + k0off;
    const long long kb0 = (long long)(2 * half) * Bsk;
    const float* bp0 = Bb + (long long)(n0 +  0 + l16) * Bsn + kb0;
    const float* bp1 = Bb + (long long)(n0 + 16 + l16) * Bsn + kb0;
    const float* bp2 = Bb + (long long)(n0 + 32 + l16) * Bsn + kb0;
    const float* bp3 = Bb + (long long)(n0 + 48 + l16) * Bsn + kb0;
    const long long aStep = 4 * Ask;
    const long long bStep = 4 * Bsk;

    for (int k0 = 0; k0 < K; k0 += 4) {
        v2f a, b0, b1, b2, b3;
        // A 16x4 fragment: lanes 0-15 hold K={0,1}, lanes 16-31 hold K={2,3}
        a.x = ap[0];
        a.y = ap[Ask];
        // B 4x16 fragments for four adjacent N tiles (mirrored striping)
        b0.x = bp0[0];  b0.y = bp0[Bsk];
        b1.x = bp1[0];  b1.y = bp1[Bsk];
        b2.x = bp2[0];  b2.y = bp2[Bsk];
        b3.x = bp3[0];  b3.y = bp3[Bsk];
        acc0 = __builtin_amdgcn_wmma_f32_16x16x4_f32(false, a, false, b0,
                                                     (short)0, acc0, false, false);
        acc1 = __builtin_amdgcn_wmma_f32_16x16x4_f32(false, a, false, b1,
                                                     (short)0, acc1, false, false);
        acc2 = __builtin_amdgcn_wmma_f32_16x16x4_f32(false, a, false, b2,
                                                     (short)0, acc2, false, false);
        acc3 = __builtin_amdgcn_wmma_f32_16x16x4_f32(false, a, false, b3,
                                                     (short)0, acc3, false, false);
        ap  += aStep;
        bp0 += bStep; bp1 += bStep; bp2 += bStep; bp3 += bStep;
    }

    // C/D 16x16 layout: VGPR r -> row m0+r (lanes 0-15) / m0+8+r (lanes 16-31)
    const int m_base = m0 + 8 * half;
    const float bm = (biasMode == 1) ? bias[m_base] : 0.f; // per-row handled below
#pragma unroll
    for (int r = 0; r < 8; ++r) {
        const int m = m0 + r + 8 * half;
        float rowBias = 0.f;
        if (biasMode == 1) rowBias = bias[m];
        float* crow = Cb + (long long)m * Csm;
#pragma unroll
        for (int t = 0; t < 4; ++t) {
            const int n = n0 + 16 * t + l16;
            float v;
            if (t == 0)      v = acc0[r];
            else if (t == 1) v = acc1[r];
            else if (t == 2) v = acc2[r];
            else             v = acc3[r];
            v = v * alpha + rowBias;
            if (biasMode == 2) v += bias[n];
            crow[(long long)n * Csn] = v;
        }
    }
    (void)bm;
}

// ---------------------------------------------------------------------------
// Farthest point sampling (matches jax scan: cent[i] = far before update,
// argmax takes the first maximum). One block per batch. Writes new_xyz
// directly into d_out in (B,3,S) layout.
// ---------------------------------------------------------------------------
__global__ void k_fps(const float* __restrict__ xyz, int* __restrict__ fpsidx,
                      float* __restrict__ out_newxyz)
{
    __shared__ float px[NN], py[NN], pz[NN];
    __shared__ float rv[256];
    __shared__ int   ri[256];
    __shared__ int   s_far;

    const int b = blockIdx.x;
    const int t = threadIdx.x;
    const float* base = xyz + (long long)b * 3 * NN;

    for (int n = t; n < NN; n += 256) {
        px[n] = base[n];
        py[n] = base[NN + n];
        pz[n] = base[2 * NN + n];
    }
    float dist[NN / 256];
#pragma unroll
    for (int j = 0; j < NN / 256; ++j) dist[j] = 1e10f;
    if (t == 0) s_far = 0;
    __syncthreads();

    for (int i = 0; i < SS; ++i) {
        const int far = s_far;
        if (t == 0) fpsidx[b * SS + i] = far;
        const float cx = px[far], cy = py[far], cz = pz[far];
        float bv = -1.f; int bi = 0;
#pragma unroll
        for (int j = 0; j < NN / 256; ++j) {
            const int n = t + j * 256;           // ascending within thread
            const float dx = px[n] - cx, dy = py[n] - cy, dz = pz[n] - cz;
            const float d  = dx * dx + dy * dy + dz * dz;
            const float dm = fminf(dist[j], d);
            dist[j] = dm;
            if (dm > bv) { bv = dm; bi = n; }    // strict >: keep first max
        }
        rv[t] = bv; ri[t] = bi;
        __syncthreads();
        for (int s2 = 128; s2 > 0; s2 >>= 1) {
            if (t < s2) {
                const float v2 = rv[t + s2]; const int i2 = ri[t + s2];
                if (v2 > rv[t] || (v2 == rv[t] && i2 < ri[t])) { rv[t] = v2; ri[t] = i2; }
            }
            __syncthreads();
        }
        if (t == 0) s_far = ri[0];
        __syncthreads();
    }

    for (int s = t; s < SS; s += 256) {
        const int id = fpsidx[b * SS + s];
        out_newxyz[(long long)b * 3 * SS + 0 * SS + s] = px[id];
        out_newxyz[(long long)b * 3 * SS + 1 * SS + s] = py[id];
        out_newxyz[(long long)b * 3 * SS + 2 * SS + s] = pz[id];
    }
}

// gather new_f into (B, CIN, S) layout ("inputs" for branch 1, also used as mean)
__global__ void k_gather_newf(const float* __restrict__ feat,
                              const int* __restrict__ fpsidx,
                              float* __restrict__ inputs)
{
    const long long i = (long long)blockIdx.x * blockDim.x + threadIdx.x;
    if (i >= (long long)BB * CINC * SS) return;
    const int s = (int)(i % SS);
    const int c = (int)((i / SS) % CINC);
    const int b = (int)(i / ((long long)CINC * SS));
    inputs[i] = feat[((long long)b * CINC + c) * NN + fpsidx[b * SS + s]];
}

// ---------------------------------------------------------------------------
// Top-K=32 nearest neighbors by sqdist (|a|^2+|b|^2-2ab form like reference);
// masked argmin passes with first-index tie-break (matches stable top_k).
// ---------------------------------------------------------------------------
__global__ void k_topk(const float* __restrict__ xyz,
                       const float* __restrict__ newxyz, // d_out (B,3,S)
                       int* __restrict__ knn)
{
    __shared__ float d[NN];
    __shared__ float rv[128];
    __shared__ int   ri[128];
    const int s = blockIdx.x, b = blockIdx.y, t = threadIdx.x;
    const float* base = xyz + (long long)b * 3 * NN;
    const float cx = newxyz[(long long)b * 3 * SS + s];
    const float cy = newxyz[(long long)b * 3 * SS + SS + s];
    const float cz = newxyz[(long long)b * 3 * SS + 2 * SS + s];
    const float c2 = cx * cx + cy * cy + cz * cz;

    for (int n = t; n < NN; n += 128) {
        const float x = base[n], y = base[NN + n], z = base[2 * NN + n];
        d[n] = c2 + x * x + y * y + z * z - 2.f * (cx * x + cy * y + cz * z);
    }
    __syncthreads();

    for (int k = 0; k < KK; ++k) {
        float bv = 1e30f; int bi = 0;
#pragma unroll
        for (int j = 0; j < NN / 128; ++j) {
            const int n = t + j * 128;           // ascending within thread
            const float v = d[n];
            if (v < bv) { bv = v; bi = n; }      // strict <: keep first min
        }
        rv[t] = bv; ri[t] = bi;
        __syncthreads();
        for (int s2 = 64; s2 > 0; s2 >>= 1) {
            if (t < s2) {
                const float v2 = rv[t + s2]; const int i2 = ri[t + s2];
                if (v2 < rv[t] || (v2 == rv[t] && i2 < ri[t])) { rv[t] = v2; ri[t] = i2; }
            }
            __syncthreads();
        }
        if (t == 0) {
            knn[(((long long)b * SS + s) << 5) + k] = ri[0];
            d[ri[0]] = 1e30f;
        }
        __syncthreads();
    }
}

// per-batch sum / sumsq of (grouped - mean) for the ddof=1 std
__global__ void k_group_stats(const float* __restrict__ feat,
                              const float* __restrict__ newf,
                              const int* __restrict__ knn,
                              float* __restrict__ acc)
{
    __shared__ float ss[256], sq[256];
    const int s = blockIdx.x, b = blockIdx.y, t = threadIdx.x;
    float lsum = 0.f, lsq = 0.f;
    for (int e = t; e < KK * CINC; e += 256) {
        const int k = e >> 6, c = e & 63;
        const int nbr = knn[(((long long)b * SS + s) << 5) + k];
        const float g = feat[((long long)b * CINC + c) * NN + nbr];
        const float diff = g - newf[((long long)b * CINC + c) * SS + s];
        lsum += diff; lsq += diff * diff;
    }
    ss[t] = lsum; sq[t] = lsq;
    __syncthreads();
    for (int s2 = 128; s2 > 0; s2 >>= 1) {
        if (t < s2) { ss[t] += ss[t + s2]; sq[t] += sq[t + s2]; }
        __syncthreads();
    }
    if (t == 0) { atomicAdd(&acc[b], ss[0]); atomicAdd(&acc[BB + b], sq[0]); }
}

__global__ void k_group_std(const float* __restrict__ acc, float* __restrict__ dinv)
{
    const int b = threadIdx.x;
    if (b >= BB) return;
    const float M = (float)((long long)SS * KK * CINC); // 1048576
    const float s = acc[b], q = acc[BB + b];
    const float var = (q - s * s / M) / (M - 1.f);
    dinv[b] = 1.f / (sqrtf(fmaxf(var, 0.f)) + 1e-5f);
}

// build point_feats (B, 2*CIN, S, K)
__global__ void k_point_feats(const float* __restrict__ feat,
                              const float* __restrict__ newf,
                              const int* __restrict__ knn,
                              const float* __restrict__ dinv,
                              const float* __restrict__ alpha_p,
                              const float* __restrict__ beta_p,
                              float* __restrict__ pf)
{
    const long long i = (long long)blockIdx.x * blockDim.x + threadIdx.x;
    const long long tot = (long long)BB * (2 * CINC) * SS * KK;
    if (i >= tot) return;
    const int k = (int)(i & (KK - 1));
    long long r = i >> 5;
    const int s = (int)(r & (SS - 1)); r >>= 9;
    const int c = (int)(r & 127);
    const int b = (int)(r >> 7);
    const int c2 = c & (CINC - 1);
    const int nbr = knn[(((long long)b * SS + s) << 5) + k];
    const float g  = feat[((long long)b * CINC + c2) * NN + nbr];
    const float nf = newf[((long long)b * CINC + c2) * SS + s];
    float val = alpha_p[c2] * ((g - nf) * dinv[b]) + beta_p[c2];
    if (c >= CINC) val -= nf;
    pf[i] = val;
}

// ---------------------------------------------------------------------------
// BatchNorm: per-channel mean/rstd over (B, L); one block per channel.
// ---------------------------------------------------------------------------
__global__ void k_bn_stats(const float* __restrict__ y, float* __restrict__ mean,
                           float* __restrict__ rstd, int C, int L)
{
    __shared__ float ss[256], sq[256];
    const int c = blockIdx.x, t = threadIdx.x;
    const long long tot = (long long)BB * L;
    float ls = 0.f, lq = 0.f;
    for (long long i = t; i < tot; i += 256) {
        const int b = (int)(i / L);
        const int l = (int)(i % L);
        const float v = y[((long long)b * C + c) * L + l];
        ls += v; lq += v * v;
    }
    ss[t] = ls; sq[t] = lq;
    __syncthreads();
    for (int s2 = 128; s2 > 0; s2 >>= 1) {
        if (t < s2) { ss[t] += ss[t + s2]; sq[t] += sq[t + s2]; }
        __syncthreads();
    }
    if (t == 0) {
        const float m = ss[0] / (float)tot;
        const float var = sq[0] / (float)tot - m * m;
        mean[c] = m;
        rstd[c] = rsqrtf(var + 1e-5f);
    }
}

// BN apply: optional residual add and relu; dst may be a wider-channel tensor.
__global__ void k_bn_apply(const float* __restrict__ y, float* __restrict__ dst,
                           const float* __restrict__ res,
                           const float* __restrict__ g, const float* __restrict__ be,
                           const float* __restrict__ mean, const float* __restrict__ rstd,
                           int C, int L, int dstC, int dstOff, int doRelu)
{
    const long long i = (long long)blockIdx.x * blockDim.x + threadIdx.x;
    const long long tot = (long long)BB * C * L;
    if (i >= tot) return;
    const int l = (int)(i % L);
    const int c = (int)((i / L) % C);
    const int b = (int)(i / ((long long)C * L));
    float v = (y[i] - mean[c]) * rstd[c] * g[c] + be[c];
    if (res) v += res[i];
    if (doRelu) v = fmaxf(v, 0.f);
    dst[((long long)b * dstC + (dstOff + c)) * L + l] = v;
}

// BN apply + relu + max over K for post2; writes into xcat channels [128,256)
__global__ void k_bn_apply_maxk(const float* __restrict__ y, float* __restrict__ xcat,
                                const float* __restrict__ g, const float* __restrict__ be,
                                const float* __restrict__ mean, const float* __restrict__ rstd)
{
    const long long i = (long long)blockIdx.x * blockDim.x + threadIdx.x;
    if (i >= (long long)BB * COUTC * SS) return;
    const int s = (int)(i % SS);
    const int c = (int)((i / SS) % COUTC);
    const int b = (int)(i / ((long long)COUTC * SS));
    const float* p = y + ((long long)b * COUTC + c) * ((long long)SS * KK) + (long long)s * KK;
    const float sc = rstd[c] * g[c];
    float m = -1e30f;
#pragma unroll
    for (int k = 0; k < KK; ++k) {
        float v = (p[k] - mean[c]) * sc + be[c];
        v = fmaxf(v, 0.f);
        m = fmaxf(m, v);
    }
    xcat[((long long)b * (2 * COUTC) + (COUTC + c)) * SS + s] = m;
}

__global__ void k_add_relu(const float* __restrict__ a, const float* __restrict__ b,
                           float* __restrict__ dst, long long n)
{
    const long long i = (long long)blockIdx.x * blockDim.x + threadIdx.x;
    if (i < n) dst[i] = fmaxf(a[i] + b[i], 0.f);
}

// softmax over rows of length S (attention)
__global__ void k_softmax(float* __restrict__ attn)
{
    __shared__ float red[128];
    const int row = blockIdx.x, t = threadIdx.x;
    float* p = attn + (long long)row * SS;
    float mx = -1e30f;
    for (int j = t; j < SS; j += 128) mx = fmaxf(mx, p[j]);
    red[t] = mx; __syncthreads();
    for (int s2 = 64; s2 > 0; s2 >>= 1) {
        if (t < s2) red[t] = fmaxf(red[t], red[t + s2]);
        __syncthreads();
    }
    mx = red[0];
    __syncthreads();
    float sum = 0.f;
    for (int j = t; j < SS; j += 128) { const float e = expf(p[j] - mx); p[j] = e; sum += e; }
    red[t] = sum; __syncthreads();
    for (int s2 = 64; s2 > 0; s2 >>= 1) {
        if (t < s2) red[t] += red[t + s2];
        __syncthreads();
    }
    const float inv = 1.f / red[0];
    for (int j = t; j < SS; j += 128) p[j] *= inv;
}

__global__ void k_zero(float* __restrict__ p, int n)
{
    const int i = blockIdx.x * blockDim.x + threadIdx.x;
    if (i < n) p[i] = 0.f;
}

// ---------------------------------------------------------------------------
// Host orchestration
// ---------------------------------------------------------------------------
extern "C" void kernel_launch(void* const* d_in, const int* in_sizes, int n_in,
                              void* d_out, int out_size, void* d_ws, size_t ws_size,
                              hipStream_t stream)
{
    (void)in_sizes; (void)n_in; (void)out_size; (void)ws_size;

    const float* xyz     = (const float*)d_in[0];
    const float* feat    = (const float*)d_in[1];
    const float* alpha_p = (const float*)d_in[2];
    const float* beta_p  = (const float*)d_in[3];
    // layers: 0 pre1, 1 k, 2 q, 3 v, 4 post1, 5 r1a, 6 r1b, 7 r2a, 8 r2b,
    //         9 pre2, 10 post2, 11 fin ; each has _w,_b,_g,_beta at 4+4*L+{0..3}
    const float* mha_in_w  = (const float*)d_in[52];
    const float* mha_in_b  = (const float*)d_in[53];
    const float* mha_out_w = (const float*)d_in[54];
    const float* mha_out_b = (const float*)d_in[55];

    float* out = (float*)d_out;
    float* newxyz_out = out;                          // (B,3,S)
    float* xout       = out + (long long)BB * 3 * SS; // (B,128,S)

    // ---- workspace carve-up (floats) ----
    float* w = (float*)d_ws;
    size_t off = 0;
    auto alloc = [&](size_t n) { float* p = w + off; off += n; return p; };
    int*   fpsidx = (int*)alloc(BB * SS);                       // 8K
    int*   knn    = (int*)alloc((size_t)BB * SS * KK);          // 256K
    float* acc    = alloc(2 * BB);
    float* dinv   = alloc(BB);
    float* mean   = alloc(256);
    float* rstd   = alloc(256);
    float* inputs = alloc((size_t)BB * CINC * SS);              // new_f (B,64,S)
    float* resx   = alloc((size_t)BB * EE * SS);
    float* valb   = alloc((size_t)BB * EE * SS);
    float* keyb   = alloc((size_t)BB * EE * SS);
    float* qryb   = alloc((size_t)BB * EE * SS);
    float* qp     = alloc((size_t)BB * SS * EE);                // (S,E)
    float* kpT    = alloc((size_t)BB * EE * SS);                // (E,S)
    float* vp     = alloc((size_t)BB * SS * EE);                // (S,E)
    float* attn   = alloc((size_t)BB * SS * SS);                // 16 MB
    float* osd    = alloc((size_t)BB * SS * EE);                // (S,E)
    float* xat    = alloc((size_t)BB * EE * SS);                // (E,S)
    float* ta1    = alloc((size_t)BB * EE * SS);
    float* tb1    = alloc((size_t)BB * EE * SS);
    float* xcat   = alloc((size_t)BB * 2 * COUTC * SS);         // (B,256,S)
    float* yfin   = alloc((size_t)BB * COUTC * SS);             // conv scratch (128ch,S)
    const long long L2 = (long long)SS * KK;                    // 16384
    float* x1     = alloc((size_t)BB * EE * L2);                // 67 MB
    float* ta     = alloc((size_t)BB * EE * L2);
    float* tb     = alloc((size_t)BB * EE * L2);
    float* pf     = alloc((size_t)BB * 2 * CINC * L2);          // 134 MB, reused as post2 out

    // ---- GEMM launcher helpers (N must be a multiple of 64) ----
    auto gemm = [&](const float* A, const float* Bm, const float* bias, float* C,
                    int M, int N, int K,
                    long long Abat, long long Asm, long long Ask,
                    long long Bbat, long long Bsk, long long Bsn,
                    long long Cbat, long long Csm, long long Csn,
                    int biasMode, float al) {
        k_gemm<<<dim3(N / 64, M / 16, BB), 32, 0, stream>>>(
            A, Bm, bias, C, M, N, K, Abat, Asm, Ask, Bbat, Bsk, Bsn,
            Cbat, Csm, Csn, biasMode, al);
    };
    // conv(+bias)+bn(+optional residual/relu):  Y = W(OxCi) * X(B,Ci,L)
    auto convbn = [&](int layer, const float* X, float* Y, int O, int Ci, int L,
                      float* dst, int dstC, int dstOff, int relu, const float* res) {
        const float* Wt   = (const float*)d_in[4 + 4 * layer + 0];
        const float* bias = (const float*)d_in[4 + 4 * layer + 1];
        const float* g    = (const float*)d_in[4 + 4 * layer + 2];
        const float* be   = (const float*)d_in[4 + 4 * layer + 3];
        gemm(Wt, X, bias, Y, O, L, Ci,
             0, Ci, 1,
             (long long)Ci * L, L, 1,
             (long long)O * L, L, 1,
             1, 1.0f);
        k_bn_stats<<<O, 256, 0, stream>>>(Y, mean, rstd, O, L);
        const long long tot = (long long)BB * O * L;
        k_bn_apply<<<(unsigned)((tot + 255) / 256), 256, 0, stream>>>(
            Y, dst, res, g, be, mean, rstd, O, L, dstC, dstOff, relu);
    };

    // ================= sampling / grouping =================
    k_zero<<<1, 64, 0, stream>>>(acc, 2 * BB);
    k_fps<<<BB, 256, 0, stream>>>(xyz, fpsidx, newxyz_out);
    {
        const long long tot = (long long)BB * CINC * SS;
        k_gather_newf<<<(unsigned)((tot + 255) / 256), 256, 0, stream>>>(feat, fpsidx, inputs);
    }
    k_topk<<<dim3(SS, BB), 128, 0, stream>>>(xyz, newxyz_out, knn);
    k_group_stats<<<dim3(SS, BB), 256, 0, stream>>>(feat, inputs, knn, acc);
    k_group_std<<<1, 32, 0, stream>>>(acc, dinv);
    {
        const long long tot = (long long)BB * 2 * CINC * SS * KK;
        k_point_feats<<<(unsigned)((tot + 255) / 256), 256, 0, stream>>>(
            feat, inputs, knn, dinv, alpha_p, beta_p, pf);
    }

    // ================= branch 1: attention over (E,S) =================
    convbn(0, inputs, resx, EE, CINC, SS, resx, EE, 0, 1, nullptr);  // pre1
    convbn(3, resx, valb, EE, EE, SS, valb, EE, 0, 1, nullptr);      // v
    convbn(1, resx, keyb, EE, EE, SS, keyb, EE, 0, 1, nullptr);      // k
    convbn(2, resx, qryb, EE, EE, SS, qryb, EE, 0, 1, nullptr);      // q

    const long long ES = (long long)EE * SS;
    // qp(S,E) = query^T * Wq^T + bq
    gemm(qryb, mha_in_w, mha_in_b, qp, SS, EE, EE,
         ES, 1, SS,   0, 1, EE,   ES, EE, 1,   2, 1.0f);
    // kpT(E,S)
    gemm(keyb, mha_in_w + EE * EE, mha_in_b + EE, kpT, SS, EE, EE,
         ES, 1, SS,   0, 1, EE,   ES, 1, SS,   2, 1.0f);
    // vp(S,E)
    gemm(valb, mha_in_w + 2 * EE * EE, mha_in_b + 2 * EE, vp, SS, EE, EE,
         ES, 1, SS,   0, 1, EE,   ES, EE, 1,   2, 1.0f);
    // attn = (qp x kpT) / sqrt(E)
    gemm(qp, kpT, nullptr, attn, SS, SS, EE,
         ES, EE, 1,   ES, SS, 1,   (long long)SS * SS, SS, 1,   0, 0.125f);
    k_softmax<<<BB * SS, 128, 0, stream>>>(attn);
    // o(S,E) = attn x vp
    gemm(attn, vp, nullptr, osd, SS, EE, SS,
         (long long)SS * SS, SS, 1,   ES, EE, 1,   ES, EE, 1,   0, 1.0f);
    // out-proj, stored transposed as (E,S)
    gemm(osd, mha_out_w, mha_out_b, xat, SS, EE, EE,
         ES, EE, 1,   0, 1, EE,   ES, 1, SS,   2, 1.0f);
    k_add_relu<<<(unsigned)((BB * ES + 255) / 256), 256, 0, stream>>>(
        xat, resx, xat, (long long)BB * ES);                    // x = relu(o + res_x)

    convbn(5, xat, ta1, EE, EE, SS, ta1, EE, 0, 0, nullptr);        // r1a (no relu)
    convbn(6, ta1, tb1, EE, EE, SS, xat, EE, 0, 1, xat);            // r1b + res + relu
    convbn(4, xat, yfin, COUTC, EE, SS, xcat, 2 * COUTC, 0, 1, nullptr); // post1 -> xcat[0:128)

    // ================= branch 2: grouped convs over L2 = S*K =================
    convbn(9, pf, x1, EE, 2 * CINC, (int)L2, x1, EE, 0, 1, nullptr); // pre2
    convbn(7, x1, ta, EE, EE, (int)L2, ta, EE, 0, 0, nullptr);       // r2a (no relu)
    convbn(8, ta, tb, EE, EE, (int)L2, x1, EE, 0, 1, x1);            // r2b + res + relu
    {   // post2: conv into pf (same size), bn stats, then fused relu+max over K
        const float* Wt   = (const float*)d_in[4 + 4 * 10 + 0];
        const float* bias = (const float*)d_in[4 + 4 * 10 + 1];
        const float* g    = (const float*)d_in[4 + 4 * 10 + 2];
        const float* be   = (const float*)d_in[4 + 4 * 10 + 3];
        gemm(Wt, x1, bias, pf, COUTC, (int)L2, EE,
             0, EE, 1,   (long long)EE * L2, L2, 1,   (long long)COUTC * L2, L2, 1,
             1, 1.0f);
        k_bn_stats<<<COUTC, 256, 0, stream>>>(pf, mean, rstd, COUTC, (int)L2);
        const long long tot = (long long)BB * COUTC * SS;
        k_bn_apply_maxk<<<(unsigned)((tot + 255) / 256), 256, 0, stream>>>(
            pf, xcat, g, be, mean, rstd);                            // -> xcat[128:256)
    }

    // ================= final fusion =================
    convbn(11, xcat, yfin, COUTC, 2 * COUTC, SS, xout, COUTC, 0, 1, nullptr); // fin -> d_out
}